// UEmbedding_83167746720087
// MI455X (gfx1250) — compile-verified
//
#include <hip/hip_runtime.h>
#include <math.h>
#include <stdint.h>

// ---- problem constants (from reference) ----
#define Bb   32
#define Oo   64
#define FdK  512
#define Cc   512
#define Ll   16
#define SQ   256          // 16*16
#define D2   512
#define Nn   576          // Cc + Oo
#define BN   (Bb*Nn)
#define NG   8            // node-groups for split Nbar reduction (576 = 8*72)
#define GSZ  72
#define LDSTR 516         // 16x512 tile row stride, pad to kill LDS bank conflicts

#define HAVE_WMMA __has_builtin(__builtin_amdgcn_wmma_f32_16x16x4_f32)

typedef __attribute__((ext_vector_type(2))) float v2f;
typedef __attribute__((ext_vector_type(8))) float v8f;

// D = X(16x16) @ Y(16x16), f32, via 4x V_WMMA_F32_16X16X4_F32.
// TA: use X^T as A;  TB: use Y^T as B.  Result in ISA C/D layout.
template<bool TA, bool TB>
__device__ __forceinline__ v8f wmma16_ld(const float* X, int ldx, const float* Y, int ldy) {
  const int lane = threadIdx.x & 31, half = lane >> 4, q16 = lane & 15;
  v8f c = {};
#if HAVE_WMMA
  #pragma unroll
  for (int q = 0; q < 4; ++q) {
    const int k0 = q * 4 + 2 * half;
    v2f a, b;
    a.x = TA ? X[k0 * ldx + q16]       : X[q16 * ldx + k0];
    a.y = TA ? X[(k0 + 1) * ldx + q16] : X[q16 * ldx + k0 + 1];
    b.x = TB ? Y[q16 * ldy + k0]       : Y[k0 * ldy + q16];
    b.y = TB ? Y[q16 * ldy + k0 + 1]   : Y[(k0 + 1) * ldy + q16];
    c = __builtin_amdgcn_wmma_f32_16x16x4_f32(false, a, false, b, (short)0, c, false, false);
  }
#else
  #pragma unroll
  for (int r = 0; r < 8; ++r) {
    const int m = r + 8 * half;
    float s = 0.f;
    for (int k = 0; k < 16; ++k) {
      float xa = TA ? X[k * ldx + m]   : X[m * ldx + k];
      float yb = TB ? Y[q16 * ldy + k] : Y[k * ldy + q16];
      s += xa * yb;
    }
    c[r] = s;
  }
#endif
  return c;
}

__device__ __forceinline__ void storeC(float* D, int ldd, v8f c) {
  const int lane = threadIdx.x & 31, half = lane >> 4, n = lane & 15;
  #pragma unroll
  for (int r = 0; r < 8; ++r) D[(r + 8 * half) * ldd + n] = c[r];
}

__device__ __forceinline__ float actf(float x) {
  return fminf(fmaxf(x, -1.f), 5.f) + 0.1f * x;
}

// ---------------- kernels ----------------

__global__ void k_init_att(float* att) {
  int i = blockIdx.x * blockDim.x + threadIdx.x;
  if (i < BN) att[i] = 1.0f;
}

// WT[f,d] = W[d,f]  (coalesced writes; one-time 1MB)
__global__ void k_transpose(const float* __restrict__ W, float* __restrict__ WT) {
  int i = blockIdx.x * 256 + threadIdx.x;   // 512*512 total
  int f = i >> 9, d = i & 511;
  WT[f * D2 + d] = W[d * FdK + f];
}

// concept nodes: cn[c] = ct[c,:256].as16x16 @ ct[c,256:].as16x16   (batch-invariant)
__global__ __launch_bounds__(256) void k_cnodes(const float* __restrict__ ct, float* __restrict__ cn) {
  const int w = threadIdx.x >> 5;
  const int c = blockIdx.x * 8 + w;
  const float* X = ct + (size_t)c * D2;
  v8f r = wmma16_ld<false, false>(X, 16, X + SQ, 16);
  storeC(cn + (size_t)c * SQ, 16, r);
}

// fused: objects = scene @ feat_W^T + feat_b, then onode = row[:256] @ row[256:]
// one block = 16 (b,o) rows; scene tile staged via ASYNC global->LDS DMA path.
__global__ __launch_bounds__(256) void k_objnodes(const float* __restrict__ scene,
                                                  const float* __restrict__ WT,
                                                  const float* __restrict__ fbias,
                                                  float* __restrict__ on) {
  __shared__ float sA[16 * LDSTR];
  __shared__ float sO[16 * LDSTR];
  const int tid = threadIdx.x;
  const int tile = blockIdx.x;                       // 0..127
  // ---- async stage 16x512 scene tile into LDS (GLOBAL_LOAD_ASYNC_TO_LDS_B128) ----
  #pragma unroll
  for (int i = 0; i < 8; ++i) {
    const int f4  = tid + i * 256;                   // float4 index, 0..2047
    const int row = f4 >> 7;
    const int col = (f4 * 4) & 511;
    const float* gp = &scene[((size_t)tile * 16 + row) * FdK + col];
    const uint32_t loff = (uint32_t)(uintptr_t)(&sA[row * LDSTR + col]); // LDS byte offset
    unsigned long long ga = (unsigned long long)(uintptr_t)gp;
    asm volatile("global_load_async_to_lds_b128 %0, %1, off"
                 :: "v"(loff), "v"(ga) : "memory");
  }
  asm volatile("s_wait_asynccnt 0x0" ::: "memory");
  __syncthreads();
  const int w = tid >> 5, lane = tid & 31, half = lane >> 4, q16 = lane & 15;
  for (int i = 0; i < 4; ++i) {
    const int n0 = (w + 8 * i) * 16;
    v8f acc = {};
#if HAVE_WMMA
    #pragma unroll 4
    for (int kq = 0; kq < 128; ++kq) {
      const int k0 = kq * 4 + 2 * half;
      v2f a, b;
      a.x = sA[q16 * LDSTR + k0];
      a.y = sA[q16 * LDSTR + k0 + 1];
      b.x = WT[(size_t)k0 * D2 + n0 + q16];
      b.y = WT[(size_t)(k0 + 1) * D2 + n0 + q16];
      acc = __builtin_amdgcn_wmma_f32_16x16x4_f32(false, a, false, b, (short)0, acc, false, false);
    }
#else
    for (int k = 0; k < FdK; ++k) {
      float yb = WT[(size_t)k * D2 + n0 + q16];
      #pragma unroll
      for (int r = 0; r < 8; ++r) acc[r] += sA[(r + 8 * half) * LDSTR + k] * yb;
    }
#endif
    const float bias = fbias[n0 + q16];
    #pragma unroll
    for (int r = 0; r < 8; ++r) sO[(r + 8 * half) * LDSTR + n0 + q16] = acc[r] + bias;
  }
  __syncthreads();
  for (int rr = w; rr < 16; rr += 8) {               // 16x16x16 node products
    const float* row = sO + rr * LDSTR;
    v8f d = wmma16_ld<false, false>(row, 16, row + SQ, 16);
    storeC(on + ((size_t)tile * 16 + rr) * SQ, 16, d);
  }
}

// step 1/3: partial weighted node reduction, grid (B, NG) for full-chip L2 bandwidth.
// part[b][g][e] = sum_{n in group g} att[b,n] * nodes[b,n][e]   (fixed order -> deterministic)
__global__ __launch_bounds__(256) void k_stepA(const float* __restrict__ cn,
                                               const float* __restrict__ on,
                                               const float* __restrict__ att,
                                               float* __restrict__ part) {
  const int b = blockIdx.x, g = blockIdx.y, e = threadIdx.x;
  const float* attb = att + (size_t)b * Nn;
  float acc = 0.f;
  #pragma unroll 4
  for (int t = 0; t < GSZ; ++t) {
    const int n = g * GSZ + t;
    const float* row = (n < Cc) ? (cn + (size_t)n * SQ)
                                : (on + ((size_t)b * Oo + (n - Cc)) * SQ);
    __builtin_prefetch(row + SQ, 0, 1);              // next row -> global_prefetch_b8
    acc += attb[n] * row[e];
  }
  part[((size_t)b * NG + g) * SQ + e] = acc;
}

// step 2/3 (one block per b): combine partials -> Nbar; S = a1^T Nbar a2 ; Pd = a1 (cn0-cn1) a2^T
__global__ __launch_bounds__(256) void k_stepTiny(const float* __restrict__ cn,
                                                  const float* __restrict__ part,
                                                  const float* __restrict__ ct,
                                                  const int* __restrict__ pconc, int l,
                                                  float* __restrict__ Sw, float* __restrict__ Pw) {
  __shared__ float nb[SQ], cnd[SQ], t1[SQ], t2[SQ];
  const int b = blockIdx.x, e = threadIdx.x;
  float acc = 0.f;
  #pragma unroll
  for (int g = 0; g < NG; ++g) acc += part[((size_t)b * NG + g) * SQ + e];
  nb[e]  = acc;
  cnd[e] = cn[e] - cn[SQ + e];                       // nodes[0] - nodes[1]
  __syncthreads();
  const float* a1 = ct + (size_t)pconc[b * Ll + l] * D2;
  const float* a2 = a1 + SQ;
  const int w = e >> 5;
  if (w == 0)      { v8f t = wmma16_ld<true,  false>(a1, 16, nb,  16); storeC(t1, 16, t); }
  else if (w == 1) { v8f t = wmma16_ld<false, false>(a1, 16, cnd, 16); storeC(t2, 16, t); }
  __syncthreads();
  if (w == 0)      { v8f s = wmma16_ld<false, false>(t1, 16, a2, 16); storeC(Sw + (size_t)b * SQ, 16, s); }
  else if (w == 1) { v8f p = wmma16_ld<false, true >(t2, 16, a2, 16); storeC(Pw + (size_t)b * SQ, 16, p); }
}

// step 3/3: new_att[b,j] = op==0 ? act(att_j * <N_j, Pd>) : act(<N_j, S>)
// one wave per (b,j); 1KB coalesced row reads; wave32 xor-reduction. 2304 blocks.
__global__ __launch_bounds__(256) void k_stepB(const float* __restrict__ cn,
                                               const float* __restrict__ on,
                                               const float* __restrict__ Sw,
                                               const float* __restrict__ Pw,
                                               const int* __restrict__ pops, int l,
                                               float* __restrict__ att, float* __restrict__ hist) {
  const int b = blockIdx.x;
  const int w = threadIdx.x >> 5, lane = threadIdx.x & 31;
  const int j = blockIdx.y * 8 + w;
  const float* row = (j < Cc) ? (cn + (size_t)j * SQ)
                              : (on + ((size_t)b * Oo + (j - Cc)) * SQ);
  const int op = pops[b * Ll + l];
  const float* P = (op == 0) ? (Pw + (size_t)b * SQ) : (Sw + (size_t)b * SQ);
  const float4* r4 = (const float4*)(row + lane * 8);
  const float4* p4 = (const float4*)(P + lane * 8);
  float4 x0 = r4[0], x1 = r4[1], y0 = p4[0], y1 = p4[1];
  float d = x0.x * y0.x + x0.y * y0.y + x0.z * y0.z + x0.w * y0.w
          + x1.x * y1.x + x1.y * y1.y + x1.z * y1.z + x1.w * y1.w;
  #pragma unroll
  for (int off = 16; off > 0; off >>= 1) d += __shfl_xor(d, off, 32);
  const float a  = att[(size_t)b * Nn + j];
  const float x  = (op == 0) ? a * d : d;
  const float na = actf(x);
  if (lane == 0) {
    att[(size_t)b * Nn + j] = na;
    hist[(size_t)l * BN + (size_t)b * Nn + j] = na;
  }
}

__global__ __launch_bounds__(256) void k_lsm(const float* __restrict__ att, float* __restrict__ out) {
  __shared__ float red[256];
  const int b = blockIdx.x, t = threadIdx.x;
  const float* ab = att + (size_t)b * Nn;
  float m = -3.402823466e38f;
  for (int j = t; j < Nn; j += 256) m = fmaxf(m, ab[j]);
  red[t] = m; __syncthreads();
  for (int s = 128; s > 0; s >>= 1) { if (t < s) red[t] = fmaxf(red[t], red[t + s]); __syncthreads(); }
  const float M = red[0]; __syncthreads();
  float acc = 0.f;
  for (int j = t; j < Nn; j += 256) acc += expf(ab[j] - M);
  red[t] = acc; __syncthreads();
  for (int s = 128; s > 0; s >>= 1) { if (t < s) red[t] += red[t + s]; __syncthreads(); }
  const float lse = M + logf(red[0]);
  for (int j = t; j < Nn; j += 256) out[(size_t)b * Nn + j] = ab[j] - lse;
}

// ---------------- launcher ----------------

extern "C" void kernel_launch(void* const* d_in, const int* in_sizes, int n_in,
                              void* d_out, int out_size, void* d_ws, size_t ws_size,
                              hipStream_t stream) {
  const float* scene = (const float*)d_in[0];
  const int*   pops  = (const int*)  d_in[1];
  const int*   pconc = (const int*)  d_in[2];
  const float* ct    = (const float*)d_in[3];
  const float* fW    = (const float*)d_in[4];
  const float* fbias = (const float*)d_in[5];
  float* out = (float*)d_out;

  float* ws = (float*)d_ws;
  float* WT   = ws; ws += (size_t)FdK * D2;      // 1 MB transposed weights
  float* cn   = ws; ws += (size_t)Cc * SQ;       // 512 KB concept nodes (batch-shared)
  float* on   = ws; ws += (size_t)Bb * Oo * SQ;  // 2 MB object nodes
  float* att  = ws; ws += BN;
  float* part = ws; ws += (size_t)Bb * NG * SQ;  // partial Nbar sums
  float* Sw   = ws; ws += (size_t)Bb * SQ;
  float* Pw   = ws; ws += (size_t)Bb * SQ;

  k_init_att<<<(BN + 255) / 256, 256, 0, stream>>>(att);
  k_transpose<<<(FdK * D2) / 256, 256, 0, stream>>>(fW, WT);
  k_cnodes<<<Cc / 8, 256, 0, stream>>>(ct, cn);
  k_objnodes<<<(Bb * Oo) / 16, 256, 0, stream>>>(scene, WT, fbias, on);

  float* hist = out + BN;
  for (int l = 0; l < Ll; ++l) {
    k_stepA<<<dim3(Bb, NG), 256, 0, stream>>>(cn, on, att, part);
    k_stepTiny<<<Bb, 256, 0, stream>>>(cn, part, ct, pconc, l, Sw, Pw);
    k_stepB<<<dim3(Bb, Nn / 8), 256, 0, stream>>>(cn, on, Sw, Pw, pops, l, att, hist);
  }
  k_lsm<<<Bb, 256, 0, stream>>>(att, out);
}